// ColorShader_5282809774750
// MI455X (gfx1250) — compile-verified
//
#include <hip/hip_runtime.h>
#include <hip/hip_bf16.h>

// softmax_rgb_blend (pytorch3d) — memory-bound per-pixel reduction over K=10
// fragments. MI455X strategy: saturate 23.3 TB/s HBM with one thread per
// pixel, fully unrolled K loop (prob/zinv in VGPRs), coalesced streaming
// loads, v_exp/v_rcp transcendentals, global_prefetch for the colors stream,
// single b128 store per pixel. WMMA is not applicable: no shared GEMM operand.

#define KFRAG 10

__device__ __forceinline__ float fast_rcp(float x) {
    return __builtin_amdgcn_rcpf(x);
}

__global__ __launch_bounds__(256) void softmax_blend_kernel(
    const float* __restrict__ colors,        // [P, K, 3]
    const int*   __restrict__ pix_to_face,   // [P, K]
    const float* __restrict__ dists,         // [P, K]
    const float* __restrict__ zbuf,          // [P, K]
    float4*      __restrict__ out,           // [P] (rgba)
    int n_pix)
{
    constexpr float INV_SIGMA = 1.0e4f;              // 1/1e-4
    constexpr float INV_GAMMA = 1.0e4f;              // 1/1e-4
    constexpr float EPS       = 1.0e-10f;
    constexpr float ZFAR      = 100.0f;
    constexpr float INV_RANGE = 1.0f / (100.0f - 1.0f); // 1/(zfar-znear)

    int p = blockIdx.x * blockDim.x + threadIdx.x;
    if (p >= n_pix) return;

    const int*   f = pix_to_face + (size_t)p * KFRAG;
    const float* d = dists       + (size_t)p * KFRAG;
    const float* z = zbuf        + (size_t)p * KFRAG;
    const float* c = colors      + (size_t)p * (KFRAG * 3);

    // Prefetch the colors stream (used only in the second loop) while the
    // first reduction loop executes — emits global_prefetch_b8 on gfx1250.
    __builtin_prefetch(c, 0, 1);
    __builtin_prefetch(c + 16, 0, 1);

    float prob[KFRAG];
    float zinv[KFRAG];
    float one_minus = 1.0f;   // prod(1 - prob)
    float zmax = EPS;         // z_inv_max floored at EPS

#pragma unroll
    for (int k = 0; k < KFRAG; ++k) {
        int   face  = f[k];
        float dk    = d[k];
        float zk    = z[k];
        bool  valid = (face >= 0);

        // sigmoid(-d/sigma) = 1 / (1 + exp(d/sigma))
        float e  = __expf(dk * INV_SIGMA);
        float pr = fast_rcp(1.0f + e);
        pr = valid ? pr : 0.0f;

        float zi = valid ? (ZFAR - zk) * INV_RANGE : 0.0f;

        prob[k] = pr;
        zinv[k] = zi;
        zmax = fmaxf(zmax, zi);
        one_minus *= (1.0f - pr);
    }

    float sumw = 0.0f, r = 0.0f, g = 0.0f, b = 0.0f;
#pragma unroll
    for (int k = 0; k < KFRAG; ++k) {
        float w = prob[k] * __expf((zinv[k] - zmax) * INV_GAMMA);
        sumw += w;
        r = fmaf(w, c[3 * k + 0], r);
        g = fmaf(w, c[3 * k + 1], g);
        b = fmaf(w, c[3 * k + 2], b);
    }

    float delta   = fmaxf(__expf((EPS - zmax) * INV_GAMMA), EPS);
    float inv_den = fast_rcp(sumw + delta);

    float4 o;
    o.x = (r + delta) * inv_den;  // background = 1.0 per channel
    o.y = (g + delta) * inv_den;
    o.z = (b + delta) * inv_den;
    o.w = 1.0f - one_minus;
    out[p] = o;                   // 16B-aligned b128 store
}

extern "C" void kernel_launch(void* const* d_in, const int* in_sizes, int n_in,
                              void* d_out, int out_size, void* d_ws, size_t ws_size,
                              hipStream_t stream) {
    const float* colors      = (const float*)d_in[0];
    const int*   pix_to_face = (const int*)  d_in[1];
    const float* dists       = (const float*)d_in[2];
    const float* zbuf        = (const float*)d_in[3];
    float4*      out         = (float4*)d_out;

    const int n_pix = in_sizes[1] / KFRAG;  // N*H*W

    const int threads = 256;                // 8 wave32 per block
    const int blocks  = (n_pix + threads - 1) / threads;
    softmax_blend_kernel<<<blocks, threads, 0, stream>>>(
        colors, pix_to_face, dists, zbuf, out, n_pix);
}